// InfoNCELoss_79310866088477
// MI455X (gfx1250) — compile-verified
//
#include <hip/hip_runtime.h>

// CDNA5 / gfx1250 fused InfoNCE loss.
//   anchors, positives: [8192, 256] fp32  ->  scalar fp32 loss
//
// Roofline: GEMM is 34.4 GFLOP fp32, fused streaming keeps HBM traffic ~16MB
// (Pn re-reads hit the 192MB L2) -> compute bound -> V_WMMA_F32_16X16X4_F32
// (fp32-exact matrix pipe). Fixed logit bound C = 1/0.07 makes exp safe with
// no online max, so the softmax denominator is a pure accumulation over tiles.

typedef __attribute__((ext_vector_type(2))) float v2f;
typedef __attribute__((ext_vector_type(8))) float v8f;

#define N_ROWS 8192
#define DIM    256
#define INV_T  14.285714285714286f   // 1 / 0.07
#define BOUND  14.285714285714286f   // max possible logit (cos<=1): exp arg <= 0
#define EPSN   1e-8f
#define LDS_STRIDE 258               // 256 + 2 pad -> conflict-free ds_load_b64

// ---------------------------------------------------------------------------
// Kernel 1: one wave (32 lanes) per row i.
//  - L2-normalize anchors[i] and positives[i] (matches x / max(||x||, eps))
//  - diag[i] = <a_n[i], p_n[i]>   (the label logit, pre-temperature)
//  - thread (0,0) zeroes the output accumulator (stream-ordered before K2)
// ---------------------------------------------------------------------------
__global__ __launch_bounds__(256)
void infonce_normalize_diag(const float* __restrict__ A,
                            const float* __restrict__ P,
                            float* __restrict__ An, float* __restrict__ Pn,
                            float* __restrict__ Diag, float* __restrict__ out) {
    if (blockIdx.x == 0 && threadIdx.x == 0) out[0] = 0.0f;

    const int wid  = threadIdx.x >> 5;
    const int lane = threadIdx.x & 31;
    const int row  = blockIdx.x * 8 + wid;
    if (row >= N_ROWS) return;

    const float* a = A + row * DIM;
    const float* p = P + row * DIM;
    const int base = lane * 8;                    // 8 contiguous floats / lane

    float av[8], pv[8];
    float aa = 0.f, pp = 0.f, ap = 0.f;
#pragma unroll
    for (int j = 0; j < 8; ++j) {
        av[j] = a[base + j];
        pv[j] = p[base + j];
        aa = fmaf(av[j], av[j], aa);
        pp = fmaf(pv[j], pv[j], pp);
        ap = fmaf(av[j], pv[j], ap);
    }
#pragma unroll
    for (int m = 16; m >= 1; m >>= 1) {           // wave32 tree reduction
        aa += __shfl_xor(aa, m, 32);
        pp += __shfl_xor(pp, m, 32);
        ap += __shfl_xor(ap, m, 32);
    }
    const float ra = 1.0f / fmaxf(sqrtf(aa), EPSN);
    const float rp = 1.0f / fmaxf(sqrtf(pp), EPSN);

    float* an = An + row * DIM;
    float* pn = Pn + row * DIM;
#pragma unroll
    for (int j = 0; j < 8; ++j) {
        an[base + j] = av[j] * ra;
        pn[base + j] = pv[j] * rp;
    }
    if (lane == 0) Diag[row] = ap * ra * rp;
}

// ---------------------------------------------------------------------------
// Kernel 2: block = 16 anchor rows (A tile in LDS), 8 waves each stream
// 16-wide positive tiles. Per tile: 64x V_WMMA_F32_16X16X4_F32 over K=256,
// split into 4 independent accumulator chains for matrix-pipe ILP.
//
// Fragment addressing (per ISA VGPR layout tables):
//   A 16x4 (MxK): lane -> row m = lane&15, kbase = 2*(lane>>4); float2 {K,K+1}
//   B 4x16 (KxN) = transpose of a positives 16x4 chunk -> same addressing
//   C 16x16:      VGPR r, half h = lane>>4 -> M = r + 8h, N = lane&15
// ---------------------------------------------------------------------------
__global__ __launch_bounds__(256)
void infonce_main(const float* __restrict__ An, const float* __restrict__ Pn,
                  const float* __restrict__ Diag, float* __restrict__ out) {
    __shared__ float As[16 * LDS_STRIDE];
    __shared__ float rowsum[16];

    const int tid     = threadIdx.x;
    const int wid     = tid >> 5;
    const int lane    = tid & 31;
    const int rowbase = blockIdx.x * 16;

    // Cooperative load of the 16x256 anchor tile into LDS (16 KB)
    for (int idx = tid; idx < 16 * 256; idx += 256) {
        const int r = idx >> 8;
        const int c = idx & 255;
        As[r * LDS_STRIDE + c] = An[(rowbase + r) * DIM + c];
    }
    if (tid < 16) rowsum[tid] = 0.0f;
    __syncthreads();

    const int mrow = lane & 15;
    const int kb   = (lane >> 4) << 1;
    const float* Abase = &As[mrow * LDS_STRIDE + kb];

    float acc[8];
#pragma unroll
    for (int r = 0; r < 8; ++r) acc[r] = 0.0f;

    for (int ct = wid; ct < N_ROWS / 16; ct += 8) {
        const float* Bbase = Pn + (ct * 16 + mrow) * DIM + kb;

        v8f c0 = {}, c1 = {}, c2 = {}, c3 = {};
#pragma unroll
        for (int k = 0; k < 64; k += 4) {
            v2f a0 = *(const v2f*)(Abase + k);
            v2f b0 = *(const v2f*)(Bbase + k);
            c0 = __builtin_amdgcn_wmma_f32_16x16x4_f32(false, a0, false, b0,
                                                       (short)0, c0, false, false);
            v2f a1 = *(const v2f*)(Abase + 64 + k);
            v2f b1 = *(const v2f*)(Bbase + 64 + k);
            c1 = __builtin_amdgcn_wmma_f32_16x16x4_f32(false, a1, false, b1,
                                                       (short)0, c1, false, false);
            v2f a2 = *(const v2f*)(Abase + 128 + k);
            v2f b2 = *(const v2f*)(Bbase + 128 + k);
            c2 = __builtin_amdgcn_wmma_f32_16x16x4_f32(false, a2, false, b2,
                                                       (short)0, c2, false, false);
            v2f a3 = *(const v2f*)(Abase + 192 + k);
            v2f b3 = *(const v2f*)(Bbase + 192 + k);
            c3 = __builtin_amdgcn_wmma_f32_16x16x4_f32(false, a3, false, b3,
                                                       (short)0, c3, false, false);
        }
        // logits tile -> exp accumulate (exp arg <= 0 by construction)
#pragma unroll
        for (int r = 0; r < 8; ++r) {
            const float l = (c0[r] + c1[r] + c2[r] + c3[r]) * INV_T - BOUND;
            acc[r] += __expf(l);
        }
    }

    // Sum over N within each 16-lane half (C layout: N = lane&15)
#pragma unroll
    for (int r = 0; r < 8; ++r) {
#pragma unroll
        for (int m = 8; m >= 1; m >>= 1) acc[r] += __shfl_xor(acc[r], m, 32);
    }
    if ((lane & 15) == 0) {
        const int half = lane >> 4;                 // rows r + 8*half
#pragma unroll
        for (int r = 0; r < 8; ++r) atomicAdd(&rowsum[r + half * 8], acc[r]);
    }
    __syncthreads();

    // Per-row loss: log(sum) + BOUND - diag*invT ; block-sum -> global atomic
    if (wid == 0) {
        float t = 0.0f;
        if (lane < 16)
            t = __logf(rowsum[lane]) + BOUND - Diag[rowbase + lane] * INV_T;
#pragma unroll
        for (int m = 8; m >= 1; m >>= 1) t += __shfl_xor(t, m, 32);
        if (lane == 0) atomicAdd(out, t * (1.0f / (float)N_ROWS));
    }
}

// ---------------------------------------------------------------------------
extern "C" void kernel_launch(void* const* d_in, const int* in_sizes, int n_in,
                              void* d_out, int out_size, void* d_ws, size_t ws_size,
                              hipStream_t stream) {
    const float* A = (const float*)d_in[0];   // anchors   [8192,256] f32
    const float* P = (const float*)d_in[1];   // positives [8192,256] f32
    float* out = (float*)d_out;               // scalar loss

    float* An   = (float*)d_ws;               //  8 MB normalized anchors
    float* Pn   = An + (size_t)N_ROWS * DIM;  //  8 MB normalized positives
    float* Diag = Pn + (size_t)N_ROWS * DIM;  // 32 KB diagonal dots

    infonce_normalize_diag<<<N_ROWS / 8, 256, 0, stream>>>(A, P, An, Pn, Diag, out);
    infonce_main<<<N_ROWS / 16, 256, 0, stream>>>(An, Pn, Diag, out);
}